// GATLayer_51436528337335
// MI455X (gfx1250) — compile-verified
//
#include <hip/hip_runtime.h>
#include <math.h>

typedef __attribute__((ext_vector_type(2))) float v2f;
typedef __attribute__((ext_vector_type(8))) float v8f;

#define N_NODES 50000
#define DIM     64
#define HEADS   4
#define HC      256        // HEADS * DIM
#define N_EDGES 800000
#define E_TOT   850000     // + self loops
#define NEG_SLOPE 0.2f

// ---------------------------------------------------------------------------
// Kernel 1: x = embedding @ W   ([N,64] x [64,256] -> [N,256])
// One wave per 16x16 tile, 8 waves (8 col-tiles) per block.
// ---------------------------------------------------------------------------
__global__ void proj_gemm(const float* __restrict__ emb,
                          const float* __restrict__ W,
                          float* __restrict__ x) {
  const int wave = threadIdx.x >> 5;
  const int lane = threadIdx.x & 31;
  const int half = lane >> 4;
  const int l16  = lane & 15;
  const int m0 = blockIdx.x * 16;
  const int n0 = (blockIdx.y * 8 + wave) * 16;

  v8f c = {};
  const float* arow = emb + (size_t)(m0 + l16) * DIM;
#pragma unroll
  for (int k = 0; k < DIM; k += 4) {
    const int ka = k + 2 * half;
    v2f a, b;
    a.x = arow[ka + 0];
    a.y = arow[ka + 1];
    b.x = W[(size_t)(ka + 0) * HC + n0 + l16];
    b.y = W[(size_t)(ka + 1) * HC + n0 + l16];
    c = __builtin_amdgcn_wmma_f32_16x16x4_f32(false, a, false, b,
                                              (short)0, c, false, false);
  }
#pragma unroll
  for (int r = 0; r < 8; ++r)
    x[(size_t)(m0 + r + 8 * half) * HC + n0 + l16] = c[r];
}

// ---------------------------------------------------------------------------
// Kernel 2: a_src[n,h] = sum_c x[n,h,c]*att_src[h,c]  (and a_dst)
// ---------------------------------------------------------------------------
__global__ void att_dots(const float* __restrict__ x,
                         const float* __restrict__ att_src,
                         const float* __restrict__ att_dst,
                         float* __restrict__ a_src,
                         float* __restrict__ a_dst) {
  const int i = blockIdx.x * blockDim.x + threadIdx.x;
  if (i >= N_NODES * HEADS) return;
  const int n = i >> 2, h = i & 3;
  const float* xp = x + (size_t)n * HC + h * DIM;
  const float* as = att_src + h * DIM;
  const float* ad = att_dst + h * DIM;
  float s = 0.f, d = 0.f;
#pragma unroll 8
  for (int c = 0; c < DIM; ++c) {
    const float v = xp[c];
    s += v * as[c];
    d += v * ad[c];
  }
  a_src[i] = s;
  a_dst[i] = d;
}

// ---------------------------------------------------------------------------
// Kernel 3: init agg=0, amax=-inf, denom=0
// ---------------------------------------------------------------------------
__global__ void init_state(float* __restrict__ agg,
                           float* __restrict__ amax,
                           float* __restrict__ denom) {
  const size_t i = (size_t)blockIdx.x * blockDim.x + threadIdx.x;
  if (i < (size_t)N_NODES * HC) agg[i] = 0.f;
  if (i < (size_t)N_NODES * HEADS) {
    amax[i]  = -INFINITY;
    denom[i] = 0.f;
  }
}

// Order-preserving float atomic max (init value must be -inf, which it is).
__device__ inline void atomicMaxF(float* addr, float val) {
  int iv = __float_as_int(val);
  if (iv >= 0) atomicMax((int*)addr, iv);
  else         atomicMin((unsigned int*)addr, (unsigned int)iv);
}

// ---------------------------------------------------------------------------
// Kernel 4: alpha = leaky_relu(a_src[src]+a_dst[dst]); segment-max into amax
// ---------------------------------------------------------------------------
__global__ void edge_logits(const int* __restrict__ srcI,
                            const int* __restrict__ dstI,
                            const float* __restrict__ a_src,
                            const float* __restrict__ a_dst,
                            float* __restrict__ alpha,
                            float* __restrict__ amax) {
  const int e = blockIdx.x * blockDim.x + threadIdx.x;
  if (e >= E_TOT) return;
  const int s = (e < N_EDGES) ? srcI[e] : (e - N_EDGES);
  const int d = (e < N_EDGES) ? dstI[e] : (e - N_EDGES);
#pragma unroll
  for (int h = 0; h < HEADS; ++h) {
    float v = a_src[s * HEADS + h] + a_dst[d * HEADS + h];
    v = (v > 0.f) ? v : NEG_SLOPE * v;
    alpha[(size_t)e * HEADS + h] = v;
    atomicMaxF(&amax[d * HEADS + h], v);
  }
}

// ---------------------------------------------------------------------------
// Kernel 5: ex = exp(alpha - amax[dst]) (in place); denom += ex
// ---------------------------------------------------------------------------
__global__ void edge_exp(const int* __restrict__ dstI,
                         float* __restrict__ alpha,   // in: logits, out: ex
                         const float* __restrict__ amax,
                         float* __restrict__ denom) {
  const int e = blockIdx.x * blockDim.x + threadIdx.x;
  if (e >= E_TOT) return;
  const int d = (e < N_EDGES) ? dstI[e] : (e - N_EDGES);
#pragma unroll
  for (int h = 0; h < HEADS; ++h) {
    const float ex = expf(alpha[(size_t)e * HEADS + h] - amax[d * HEADS + h]);
    alpha[(size_t)e * HEADS + h] = ex;
    atomicAdd(&denom[d * HEADS + h], ex);
  }
}

// ---------------------------------------------------------------------------
// Kernel 6: agg[dst] += x[src] * coef   (one wave per edge; lane = 8 channels)
// ---------------------------------------------------------------------------
__global__ void edge_scatter(const int* __restrict__ srcI,
                             const int* __restrict__ dstI,
                             const float* __restrict__ ex,
                             const float* __restrict__ denom,
                             const float* __restrict__ x,
                             float* __restrict__ agg) {
  const int e = blockIdx.x * (blockDim.x >> 5) + (threadIdx.x >> 5);
  if (e >= E_TOT) return;
  const int lane = threadIdx.x & 31;
  const int s = (e < N_EDGES) ? srcI[e] : (e - N_EDGES);
  const int d = (e < N_EDGES) ? dstI[e] : (e - N_EDGES);
  const int h = lane >> 3;  // 8 lanes per head, 8 channels per lane
  const float coef = ex[(size_t)e * HEADS + h] / (denom[d * HEADS + h] + 1e-16f);
  const float4* xp = (const float4*)(x + (size_t)s * HC + lane * 8);
  float* ap = agg + (size_t)d * HC + lane * 8;
  const float4 v0 = xp[0], v1 = xp[1];
  atomicAdd(&ap[0], v0.x * coef);
  atomicAdd(&ap[1], v0.y * coef);
  atomicAdd(&ap[2], v0.z * coef);
  atomicAdd(&ap[3], v0.w * coef);
  atomicAdd(&ap[4], v1.x * coef);
  atomicAdd(&ap[5], v1.y * coef);
  atomicAdd(&ap[6], v1.z * coef);
  atomicAdd(&ap[7], v1.w * coef);
}

// ---------------------------------------------------------------------------
// Kernel 7: out = (agg + bias) @ dense_w + dense_b   ([N,256]x[256,64])
// ---------------------------------------------------------------------------
__global__ void dense_gemm(const float* __restrict__ agg,
                           const float* __restrict__ bias,
                           const float* __restrict__ dw,
                           const float* __restrict__ db,
                           float* __restrict__ out) {
  const int wave = threadIdx.x >> 5;   // 4 waves, 4 col-tiles of 16
  const int lane = threadIdx.x & 31;
  const int half = lane >> 4;
  const int l16  = lane & 15;
  const int m0 = blockIdx.x * 16;
  const int n0 = wave * 16;

  v8f c;
  const float db_n = db[n0 + l16];
#pragma unroll
  for (int r = 0; r < 8; ++r) c[r] = db_n;

  const float* arow = agg + (size_t)(m0 + l16) * HC;
#pragma unroll 4
  for (int k = 0; k < HC; k += 4) {
    const int ka = k + 2 * half;
    v2f a, b;
    a.x = arow[ka + 0] + bias[ka + 0];
    a.y = arow[ka + 1] + bias[ka + 1];
    b.x = dw[(size_t)(ka + 0) * DIM + n0 + l16];
    b.y = dw[(size_t)(ka + 1) * DIM + n0 + l16];
    c = __builtin_amdgcn_wmma_f32_16x16x4_f32(false, a, false, b,
                                              (short)0, c, false, false);
  }
#pragma unroll
  for (int r = 0; r < 8; ++r)
    out[(size_t)(m0 + r + 8 * half) * DIM + n0 + l16] = c[r];
}

// ---------------------------------------------------------------------------
extern "C" void kernel_launch(void* const* d_in, const int* in_sizes, int n_in,
                              void* d_out, int out_size, void* d_ws, size_t ws_size,
                              hipStream_t stream) {
  const float* emb      = (const float*)d_in[0];
  const int*   graph    = (const int*)  d_in[1];   // [2, E]
  const float* W        = (const float*)d_in[2];
  const float* att_src  = (const float*)d_in[3];
  const float* att_dst  = (const float*)d_in[4];
  const float* bias     = (const float*)d_in[5];
  const float* dense_w  = (const float*)d_in[6];
  const float* dense_b  = (const float*)d_in[7];
  float* out = (float*)d_out;

  const int* srcI = graph;
  const int* dstI = graph + N_EDGES;

  // Workspace layout (floats), 256B-aligned regions.
  char* ws = (char*)d_ws;
  size_t off = 0;
  auto take = [&](size_t nfloats) {
    float* p = (float*)(ws + off);
    off += ((nfloats * sizeof(float)) + 255) & ~(size_t)255;
    return p;
  };
  float* x     = take((size_t)N_NODES * HC);   // projected features
  float* agg   = take((size_t)N_NODES * HC);   // aggregated messages
  float* alpha = take((size_t)E_TOT * HEADS);  // logits -> ex
  float* a_src = take((size_t)N_NODES * HEADS);
  float* a_dst = take((size_t)N_NODES * HEADS);
  float* amax  = take((size_t)N_NODES * HEADS);
  float* denom = take((size_t)N_NODES * HEADS);

  // 1) projection GEMM: grid (3125 row-tiles, 2) x 8 waves = 16 col-tiles
  proj_gemm<<<dim3(N_NODES / 16, 2), 256, 0, stream>>>(emb, W, x);

  // 2) attention dot products
  att_dots<<<(N_NODES * HEADS + 255) / 256, 256, 0, stream>>>(
      x, att_src, att_dst, a_src, a_dst);

  // 3) init agg / amax / denom
  init_state<<<((size_t)N_NODES * HC + 255) / 256, 256, 0, stream>>>(
      agg, amax, denom);

  // 4) edge logits + segment max
  edge_logits<<<(E_TOT + 255) / 256, 256, 0, stream>>>(
      srcI, dstI, a_src, a_dst, alpha, amax);

  // 5) exp + denominator
  edge_exp<<<(E_TOT + 255) / 256, 256, 0, stream>>>(dstI, alpha, amax, denom);

  // 6) weighted scatter aggregate: 8 edges (waves) per 256-thread block
  edge_scatter<<<(E_TOT + 7) / 8, 256, 0, stream>>>(
      srcI, dstI, alpha, denom, x, agg);

  // 7) output GEMM with bias fold: 4 waves = 4 col-tiles of 16
  dense_gemm<<<N_NODES / 16, 128, 0, stream>>>(agg, bias, dense_w, dense_b, out);
}